// Ndpdt_60292750901747
// MI455X (gfx1250) — compile-verified
//
#include <hip/hip_runtime.h>
#include <math.h>

#define NETS   5
#define TSIZE  512
#define TTOT   (NETS * TSIZE)
#define STEPS  50
#define DT_C   0.02f
#define WS_HDR 64      // dwords reserved at front of workspace (min/max live in [0..9])
#define MM_BLOCKS 1024

typedef unsigned int u32;
typedef __attribute__((ext_vector_type(4))) u32 v4u;
typedef __attribute__((ext_vector_type(8))) int v8i;
typedef __attribute__((ext_vector_type(4))) int v4i;
typedef __attribute__((ext_vector_type(2))) float v2f;
typedef __attribute__((ext_vector_type(8))) float v8f;

// ---- CDNA5 hardware tanh (V_TANH_F32, TRANS32 op, new on gfx1250) ----
__device__ __forceinline__ float tanh_hw(float x) {
#if __has_builtin(__builtin_amdgcn_tanhf)
    return __builtin_amdgcn_tanhf(x);
#elif __has_builtin(__builtin_amdgcn_tanh_f32)
    return __builtin_amdgcn_tanh_f32(x);
#else
    float r;
    asm("v_tanh_f32 %0, %1" : "=v"(r) : "v"(x));
    return r;
#endif
}

// ---- monotone float <-> uint mapping (so uint min/max == float min/max) ----
__device__ __forceinline__ u32 f2ord(float f) {
    u32 u = __float_as_uint(f);
    return (u & 0x80000000u) ? ~u : (u | 0x80000000u);
}
__device__ __forceinline__ float ord2f(u32 u) {
    u = (u & 0x80000000u) ? (u & 0x7FFFFFFFu) : ~u;
    return __uint_as_float(u);
}

// ---- closed-form symmetric 3x3 eigenvalues (trig method), t1>=t2>=t3 ----
__device__ __forceinline__ void eig_vals(float a00, float a01, float a02,
                                         float a11, float a12, float a22,
                                         float& t1, float& t2, float& t3) {
    float q   = (a00 + a11 + a22) * (1.0f / 3.0f);
    float b00 = a00 - q, b11 = a11 - q, b22 = a22 - q;
    float p1  = a01 * a01 + a02 * a02 + a12 * a12;
    float p2  = b00 * b00 + b11 * b11 + b22 * b22 + 2.0f * p1;
    float pp  = sqrtf(p2 * (1.0f / 6.0f));
    if (pp > 1e-12f) {
        float ip  = 1.0f / pp;
        float c00 = b00 * ip, c11 = b11 * ip, c22 = b22 * ip;
        float c01 = a01 * ip, c02 = a02 * ip, c12 = a12 * ip;
        float det = c00 * (c11 * c22 - c12 * c12)
                  - c01 * (c01 * c22 - c12 * c02)
                  + c02 * (c01 * c12 - c11 * c02);
        float r   = fminf(1.0f, fmaxf(-1.0f, 0.5f * det));
        float phi = acosf(r) * (1.0f / 3.0f);
        float tp  = 2.0f * pp;
        t1 = q + tp * cosf(phi);
        t3 = q + tp * cosf(phi + 2.0943951023931953f);   // +2*pi/3
        t2 = 3.0f * q - t1 - t3;
    } else {
        t1 = t2 = t3 = q;
    }
}

__device__ __forceinline__ void invariants(float t1, float t2, float t3, float inv[NETS]) {
    float trace = t1 + t2 + t3;
    float sumsq = t1 * t1 + t2 * t2 + t3 * t3;
    float crs   = t1 * t2 + t1 * t3 + t2 * t3;
    inv[0] = t1;
    inv[1] = t1 + t2;
    inv[2] = trace;
    inv[3] = sumsq + 2.0f * crs;
    inv[4] = sumsq - crs;
}

// Eigenvector for eigenvalue lam: largest cross product of rows of (A - lam I).
__device__ __forceinline__ void eig_vec(float a00, float a01, float a02,
                                        float a11, float a12, float a22,
                                        float lam, float v[3]) {
    float r0x = a00 - lam, r0y = a01,       r0z = a02;
    float r1x = a01,       r1y = a11 - lam, r1z = a12;
    float r2x = a02,       r2y = a12,       r2z = a22 - lam;
    float c0x = r0y * r1z - r0z * r1y, c0y = r0z * r1x - r0x * r1z, c0z = r0x * r1y - r0y * r1x;
    float c1x = r0y * r2z - r0z * r2y, c1y = r0z * r2x - r0x * r2z, c1z = r0x * r2y - r0y * r2x;
    float c2x = r1y * r2z - r1z * r2y, c2y = r1z * r2x - r1x * r2z, c2z = r1x * r2y - r1y * r2x;
    float n0 = c0x * c0x + c0y * c0y + c0z * c0z;
    float n1 = c1x * c1x + c1y * c1y + c1z * c1z;
    float n2 = c2x * c2x + c2y * c2y + c2z * c2z;
    float bx = c0x, by = c0y, bz = c0z, bn = n0;
    if (n1 > bn) { bx = c1x; by = c1y; bz = c1z; bn = n1; }
    if (n2 > bn) { bx = c2x; by = c2y; bz = c2z; bn = n2; }
    float inv = rsqrtf(fmaxf(bn, 1e-30f));
    v[0] = bx * inv; v[1] = by * inv; v[2] = bz * inv;
}

// Scalar 50-step Euler integration (used by the no-workspace fallback only).
__device__ __forceinline__ float ode50(float v, const float w1[3], const float w2[9],
                                       const float w3[3], float eb) {
#pragma unroll 1
    for (int s = 0; s < STEPS; ++s) {
        float h0 = tanh_hw(v * w1[0]);
        float h1 = tanh_hw(v * w1[1]);
        float h2 = tanh_hw(v * w1[2]);
        float g0 = tanh_hw(h0 * w2[0] + h1 * w2[3] + h2 * w2[6]);
        float g1 = tanh_hw(h0 * w2[1] + h1 * w2[4] + h2 * w2[7]);
        float g2 = tanh_hw(h0 * w2[2] + h1 * w2[5] + h2 * w2[8]);
        v += DT_C * (g0 * w3[0] + g1 * w3[1] + g2 * w3[2] + eb);
    }
    return v;
}

// ======================= Pass 0: init workspace header =======================
extern "C" __global__ void ws_init(u32* ws) {
    int t = threadIdx.x;
    if (t < NETS)               ws[t] = 0xFFFFFFFFu;          // running min (ordered)
    else if (t < 2 * NETS)      ws[t] = 0u;                    // running max (ordered)
}

// ======================= Pass A: exact per-net min/max of invariants =========
extern "C" __global__ void __launch_bounds__(256)
pass_minmax(const float* __restrict__ x, u32* __restrict__ ws, int P) {
    float mn[NETS], mx[NETS];
#pragma unroll
    for (int k = 0; k < NETS; ++k) { mn[k] = __builtin_inff(); mx[k] = -__builtin_inff(); }

    int stride = gridDim.x * 256;
    for (int p = blockIdx.x * 256 + threadIdx.x; p < P; p += stride) {
        const float* xp = x + (size_t)p * 9;
        float a00 = xp[0], a01 = xp[1], a02 = xp[2];
        float a11 = xp[4], a12 = xp[5], a22 = xp[8];
        float t1, t2, t3;
        eig_vals(a00, a01, a02, a11, a12, a22, t1, t2, t3);
        float inv[NETS];
        invariants(t1, t2, t3, inv);
#pragma unroll
        for (int k = 0; k < NETS; ++k) {
            mn[k] = fminf(mn[k], inv[k]);
            mx[k] = fmaxf(mx[k], inv[k]);
        }
    }

#pragma unroll
    for (int off = 16; off > 0; off >>= 1) {
#pragma unroll
        for (int k = 0; k < NETS; ++k) {
            mn[k] = fminf(mn[k], __shfl_xor(mn[k], off, 32));
            mx[k] = fmaxf(mx[k], __shfl_xor(mx[k], off, 32));
        }
    }

    __shared__ float wmn[8][NETS], wmx[8][NETS];
    int wid = threadIdx.x >> 5, lid = threadIdx.x & 31;
    if (lid == 0) {
#pragma unroll
        for (int k = 0; k < NETS; ++k) { wmn[wid][k] = mn[k]; wmx[wid][k] = mx[k]; }
    }
    __syncthreads();
    if (threadIdx.x < NETS) {
        int k = threadIdx.x;
        float m = wmn[0][k], M = wmx[0][k];
#pragma unroll
        for (int w = 1; w < 8; ++w) { m = fminf(m, wmn[w][k]); M = fmaxf(M, wmx[w][k]); }
        atomicMin(&ws[k],        f2ord(m));
        atomicMax(&ws[NETS + k], f2ord(M));
    }
}

// ======================= Pass B: tabulate F_k^(50) via WMMA ==================
// One wave = 16 consecutive table entries of one net, entry n in lanes {n, n+16}
// (state replicated across lane halves). Layer-2 matmul runs on
// V_WMMA_F32_16X16X4_F32 with A = all-ones (layout-invariant) and the scaled
// activations in B: D[m,n] = sum_k B[k,n] = h.W2[:,j] for column n — K-order
// irrelevant by symmetry, only the (documented) column<->lane mapping is used.
extern "C" __global__ void __launch_bounds__(256)
pass_table(const float* __restrict__ W1, const float* __restrict__ W2,
           const float* __restrict__ W3, const float* __restrict__ bb,
           u32* __restrict__ ws) {
    int wid  = threadIdx.x >> 5;
    int lid  = threadIdx.x & 31;
    int col  = lid & 15;
    bool lo  = lid < 16;
    int base = (blockIdx.x * 8 + wid) * 16;   // first entry of this wave's tile
    int k    = base / TSIZE;                  // net id (uniform per wave)
    int i    = (base % TSIZE) + col;          // table index within net

    float mn = ord2f(ws[k]);
    float mx = ord2f(ws[NETS + k]);
    float d  = (mx > mn) ? (mx - mn) * (1.0f / (TSIZE - 1)) : 0.0f;
    float v  = mn + d * (float)i;             // replicated: lanes n and n+16 equal

    // lane-half-dependent weight registers (built once, outside the loop)
    float wa  = lo ? W1[k * 3 + 0] : W1[k * 3 + 2];
    float wb  = lo ? W1[k * 3 + 1] : 0.0f;
    float sa0 = lo ? W2[k * 9 + 0] : W2[k * 9 + 6];   // W2[0][j] : W2[2][j]
    float sa1 = lo ? W2[k * 9 + 1] : W2[k * 9 + 7];
    float sa2 = lo ? W2[k * 9 + 2] : W2[k * 9 + 8];
    float sb0 = W2[k * 9 + 3], sb1 = W2[k * 9 + 4], sb2 = W2[k * 9 + 5];  // W2[1][j]
    float w30 = W3[k * 3 + 0], w31 = W3[k * 3 + 1], w32 = W3[k * 3 + 2];
    float eb  = expf(bb[k]);

    v2f ones; ones[0] = 1.0f; ones[1] = 1.0f;  // A = all-ones: layout-invariant
    v8f cz = {};                                // C = 0

#pragma unroll 1
    for (int s = 0; s < STEPS; ++s) {
        // layer 1: lower half computes h0,h1; upper half computes h2, 0
        float hA = tanh_hw(v * wa);
        float hB = tanh_hw(v * wb);
        // layer 2 pre-activations: column n's 4 K-slots = {h0*W2[0][j], h1*W2[1][j], h2*W2[2][j], 0}
        v2f b0; b0[0] = hA * sa0; b0[1] = hB * sb0;
        v2f b1; b1[0] = hA * sa1; b1[1] = hB * sb1;
        v2f b2; b2[0] = hA * sa2; b2[1] = hB * sb2;
        v8f d0 = __builtin_amdgcn_wmma_f32_16x16x4_f32(false, ones, false, b0, (short)0, cz, false, false);
        v8f d1 = __builtin_amdgcn_wmma_f32_16x16x4_f32(false, ones, false, b1, (short)0, cz, false, false);
        v8f d2 = __builtin_amdgcn_wmma_f32_16x16x4_f32(false, ones, false, b2, (short)0, cz, false, false);
        // rows of D are identical (A=ones) -> d[0] already holds column (lane%16)'s
        // value in every lane; both halves read the same value -> v stays replicated.
        float g0 = tanh_hw(d0[0]);
        float g1 = tanh_hw(d1[0]);
        float g2 = tanh_hw(d2[0]);
        v += DT_C * (g0 * w30 + g1 * w31 + g2 * w32 + eb);
    }
    if (lo) ((float*)ws)[WS_HDR + base + col] = v;
}

// ======================= Pass C: eigen + table interp + reconstruction =======
extern "C" __global__ void __launch_bounds__(256)
pass_apply(const float* __restrict__ x, const u32* __restrict__ ws,
           float* __restrict__ out, int P) {
    __shared__ float tb[TTOT];
    const float* gtbl = ((const float*)ws) + WS_HDR;

    // --- TDM prefetch of the table into LDS (Tensor Data Mover) ---
#if defined(__gfx1250__) && __has_builtin(__builtin_amdgcn_tensor_load_to_lds) && __has_builtin(__builtin_amdgcn_s_wait_tensorcnt)
    if (threadIdx.x < 32) {
        u32 lds_base = (u32)(uintptr_t)(&tb[0]);
        unsigned long long ga = (unsigned long long)(uintptr_t)gtbl;
        v4u g0;
        g0[0] = 1u;
        g0[1] = lds_base;
        g0[2] = (u32)ga;
        g0[3] = ((u32)(ga >> 32) & 0x01FFFFFFu) | 0x80000000u;
        v8i g1;
        g1[0] = 0x00020000;
        g1[1] = (int)((u32)TTOT << 16);
        g1[2] = (int)(1u << 16);
        g1[3] = (int)((u32)TTOT << 16);
        g1[4] = 1;
        g1[5] = TTOT;
        g1[6] = 0;
        g1[7] = 0;
        v4i gz = {0, 0, 0, 0};
        v8i gz8 = {0, 0, 0, 0, 0, 0, 0, 0};
        __builtin_amdgcn_tensor_load_to_lds(g0, g1, gz, gz, gz8, 0);
        __builtin_amdgcn_s_wait_tensorcnt((unsigned short)0);
    }
#endif
    __syncthreads();
    for (int j = threadIdx.x; j < TTOT; j += 256) tb[j] = gtbl[j];
    __syncthreads();

    int p = blockIdx.x * 256 + threadIdx.x;
    if (p >= P) return;

    const float* xp = x + (size_t)p * 9;
    float a00 = xp[0], a01 = xp[1], a02 = xp[2];
    float a11 = xp[4], a12 = xp[5], a22 = xp[8];

    float t1, t2, t3;
    eig_vals(a00, a01, a02, a11, a12, a22, t1, t2, t3);

    float v1[3], v2[3], v3[3];
    float p1 = a01 * a01 + a02 * a02 + a12 * a12;
    if (p1 > 1e-24f || fabsf(t1 - t3) > 1e-12f) {
        eig_vec(a00, a01, a02, a11, a12, a22, t1, v1);
        eig_vec(a00, a01, a02, a11, a12, a22, t3, v3);
        v2[0] = v3[1] * v1[2] - v3[2] * v1[1];
        v2[1] = v3[2] * v1[0] - v3[0] * v1[2];
        v2[2] = v3[0] * v1[1] - v3[1] * v1[0];
        float n2i = rsqrtf(fmaxf(v2[0] * v2[0] + v2[1] * v2[1] + v2[2] * v2[2], 1e-30f));
        v2[0] *= n2i; v2[1] *= n2i; v2[2] *= n2i;
    } else {
        v1[0] = 1.f; v1[1] = 0.f; v1[2] = 0.f;
        v2[0] = 0.f; v2[1] = 1.f; v2[2] = 0.f;
        v3[0] = 0.f; v3[1] = 0.f; v3[2] = 1.f;
    }

    float inv[NETS];
    invariants(t1, t2, t3, inv);

    float N[NETS];
#pragma unroll
    for (int k = 0; k < NETS; ++k) {
        float mn = ord2f(ws[k]);
        float mx = ord2f(ws[NETS + k]);
        float rs = (mx > mn) ? (float)(TSIZE - 1) / (mx - mn) : 0.0f;
        float t  = (inv[k] - mn) * rs;
        t = fminf(fmaxf(t, 0.0f), (float)(TSIZE - 1) - 1e-3f);
        int   i0 = (int)t;
        float f  = t - (float)i0;
        float a  = tb[k * TSIZE + i0];
        float b  = tb[k * TSIZE + i0 + 1];
        N[k] = a + f * (b - a);
    }

    float trace = t1 + t2 + t3;
    float t4 = 2.0f * N[3] * trace;
    float d1 = N[0] + N[1] + N[2] + t4 + N[4] * (2.0f * t1 - t2 - t3);
    float d2 =        N[1] + N[2] + t4 + N[4] * (2.0f * t2 - t1 - t3);
    float d3 =               N[2] + t4 + N[4] * (2.0f * t3 - t1 - t2);

    float m00 = d1 * v1[0] * v1[0] + d2 * v2[0] * v2[0] + d3 * v3[0] * v3[0];
    float m01 = d1 * v1[0] * v1[1] + d2 * v2[0] * v2[1] + d3 * v3[0] * v3[1];
    float m02 = d1 * v1[0] * v1[2] + d2 * v2[0] * v2[2] + d3 * v3[0] * v3[2];
    float m11 = d1 * v1[1] * v1[1] + d2 * v2[1] * v2[1] + d3 * v3[1] * v3[1];
    float m12 = d1 * v1[1] * v1[2] + d2 * v2[1] * v2[2] + d3 * v3[1] * v3[2];
    float m22 = d1 * v1[2] * v1[2] + d2 * v2[2] * v2[2] + d3 * v3[2] * v3[2];

    __builtin_nontemporal_store(m00, out + 0 * (size_t)P + p);
    __builtin_nontemporal_store(m01, out + 1 * (size_t)P + p);
    __builtin_nontemporal_store(m02, out + 2 * (size_t)P + p);
    __builtin_nontemporal_store(m11, out + 3 * (size_t)P + p);
    __builtin_nontemporal_store(m12, out + 4 * (size_t)P + p);
    __builtin_nontemporal_store(m22, out + 5 * (size_t)P + p);
}

// ======================= Fallback: direct (no workspace) =====================
extern "C" __global__ void __launch_bounds__(256)
dpsi_direct(const float* __restrict__ x, const float* __restrict__ W1,
            const float* __restrict__ W2, const float* __restrict__ W3,
            const float* __restrict__ bb, float* __restrict__ out, int P) {
    float w1[NETS][3], w2[NETS][9], w3[NETS][3], eb[NETS];
#pragma unroll
    for (int k = 0; k < NETS; ++k) {
        eb[k] = expf(bb[k]);
#pragma unroll
        for (int j = 0; j < 3; ++j) { w1[k][j] = W1[k * 3 + j]; w3[k][j] = W3[k * 3 + j]; }
#pragma unroll
        for (int j = 0; j < 9; ++j) w2[k][j] = W2[k * 9 + j];
    }
    int p = blockIdx.x * 256 + threadIdx.x;
    if (p >= P) return;
    const float* xp = x + (size_t)p * 9;
    float a00 = xp[0], a01 = xp[1], a02 = xp[2], a11 = xp[4], a12 = xp[5], a22 = xp[8];
    float t1, t2, t3;
    eig_vals(a00, a01, a02, a11, a12, a22, t1, t2, t3);
    float v1[3], v2[3], v3[3];
    eig_vec(a00, a01, a02, a11, a12, a22, t1, v1);
    eig_vec(a00, a01, a02, a11, a12, a22, t3, v3);
    v2[0] = v3[1] * v1[2] - v3[2] * v1[1];
    v2[1] = v3[2] * v1[0] - v3[0] * v1[2];
    v2[2] = v3[0] * v1[1] - v3[1] * v1[0];
    float n2i = rsqrtf(fmaxf(v2[0] * v2[0] + v2[1] * v2[1] + v2[2] * v2[2], 1e-30f));
    v2[0] *= n2i; v2[1] *= n2i; v2[2] *= n2i;
    float inv[NETS], N[NETS];
    invariants(t1, t2, t3, inv);
#pragma unroll
    for (int k = 0; k < NETS; ++k) N[k] = ode50(inv[k], w1[k], w2[k], w3[k], eb[k]);
    float trace = t1 + t2 + t3;
    float t4 = 2.0f * N[3] * trace;
    float d1 = N[0] + N[1] + N[2] + t4 + N[4] * (2.0f * t1 - t2 - t3);
    float d2 =        N[1] + N[2] + t4 + N[4] * (2.0f * t2 - t1 - t3);
    float d3 =               N[2] + t4 + N[4] * (2.0f * t3 - t1 - t2);
    out[0 * (size_t)P + p] = d1 * v1[0] * v1[0] + d2 * v2[0] * v2[0] + d3 * v3[0] * v3[0];
    out[1 * (size_t)P + p] = d1 * v1[0] * v1[1] + d2 * v2[0] * v2[1] + d3 * v3[0] * v3[1];
    out[2 * (size_t)P + p] = d1 * v1[0] * v1[2] + d2 * v2[0] * v2[2] + d3 * v3[0] * v3[2];
    out[3 * (size_t)P + p] = d1 * v1[1] * v1[1] + d2 * v2[1] * v2[1] + d3 * v3[1] * v3[1];
    out[4 * (size_t)P + p] = d1 * v1[1] * v1[2] + d2 * v2[1] * v2[2] + d3 * v3[1] * v3[2];
    out[5 * (size_t)P + p] = d1 * v1[2] * v1[2] + d2 * v2[2] * v2[2] + d3 * v3[2] * v3[2];
}

extern "C" void kernel_launch(void* const* d_in, const int* in_sizes, int n_in,
                              void* d_out, int out_size, void* d_ws, size_t ws_size,
                              hipStream_t stream) {
    const float* x  = (const float*)d_in[0];
    const float* W1 = (const float*)d_in[1];
    const float* W2 = (const float*)d_in[2];
    const float* W3 = (const float*)d_in[3];
    const float* b  = (const float*)d_in[4];
    float* out = (float*)d_out;

    int P = in_sizes[0] / 9;
    int gridP = (P + 255) / 256;
    int gridMM = gridP < MM_BLOCKS ? gridP : MM_BLOCKS;
    size_t need = (size_t)(WS_HDR + TTOT) * sizeof(u32);

    if (ws_size >= need) {
        u32* ws = (u32*)d_ws;
        hipLaunchKernelGGL(ws_init,     dim3(1),          dim3(32),  0, stream, ws);
        hipLaunchKernelGGL(pass_minmax, dim3(gridMM),     dim3(256), 0, stream, x, ws, P);
        hipLaunchKernelGGL(pass_table,  dim3(TTOT / 128), dim3(256), 0, stream, W1, W2, W3, b, ws);
        hipLaunchKernelGGL(pass_apply,  dim3(gridP),      dim3(256), 0, stream, x, ws, out, P);
    } else {
        hipLaunchKernelGGL(dpsi_direct, dim3(gridP),      dim3(256), 0, stream,
                           x, W1, W2, W3, b, out, P);
    }
}